// Backbone_82197084110893
// MI455X (gfx1250) — compile-verified
//
#include <hip/hip_runtime.h>
#include <cstdint>

#define N_NODES 100000
#define N_EDGES 1600000
#define MTILES  6250          // 100000 / 16 exactly

typedef float v2f __attribute__((ext_vector_type(2)));
typedef float v8f __attribute__((ext_vector_type(8)));

// ---------------------------------------------------------------------------
// Degree / normalization:  deg[n] = 1 (self loop) + sum_{e: col==n} ew[e]
//                          dinv[n] = rsqrt(deg[n])    (deg >= 1 always)
// ---------------------------------------------------------------------------
__global__ void deg_init_kernel(float* __restrict__ deg) {
    int i = blockIdx.x * blockDim.x + threadIdx.x;
    if (i < N_NODES) deg[i] = 1.0f;
}

__global__ void deg_scatter_kernel(const long long* __restrict__ ei,
                                   const float* __restrict__ ew,
                                   float* __restrict__ deg) {
    int e = blockIdx.x * blockDim.x + threadIdx.x;
    if (e < N_EDGES) {
        int c = (int)ei[N_EDGES + e];          // edge_index[1][e]
        unsafeAtomicAdd(&deg[c], ew[e]);       // hw global_atomic_add_f32
    }
}

__global__ void dinv_kernel(float* __restrict__ deg) {
    int i = blockIdx.x * blockDim.x + threadIdx.x;
    if (i < N_NODES) {
        float d = deg[i];
        deg[i] = (d > 0.0f) ? rsqrtf(d) : 0.0f;
    }
}

// ---------------------------------------------------------------------------
// Y[N, FOUT] = X[N, 64] @ W[64, FOUT] with V_WMMA_F32_16X16X4_F32.
// One wave per 16x16 output tile; K=64 -> 16 WMMA steps of K=4.
//
// f32 A 16x4 layout : lanes 0-15 M=0..15 hold {K=0 (v0), K=1 (v1)}
//                     lanes16-31 M=0..15 hold {K=2 (v0), K=3 (v1)}
// f32 B 4x16 layout : mirrored with N across lanes.
// f32 C/D 16x16     : VGPR j, lanes 0-15 -> M=j, N=lane; lanes16-31 -> M=j+8.
// ---------------------------------------------------------------------------
template <int FOUT>
__global__ void gemm_wmma_f32_kernel(const float* __restrict__ X,
                                     const float* __restrict__ W,
                                     float* __restrict__ Y) {
    constexpr int NT = FOUT / 16;
    const int wid   = blockIdx.x * (blockDim.x >> 5) + (threadIdx.x >> 5);
    const int mtile = wid / NT;
    const int ntile = wid % NT;
    if (mtile >= MTILES) return;               // wave-uniform: EXEC stays all-1s

    const int lane = threadIdx.x & 31;
    const int hi   = lane >> 4;                // 0: K pair {0,1}; 1: K pair {2,3}
    const int l16  = lane & 15;
    const int arow = mtile * 16 + l16;         // A: M index for this lane
    const int ncol = ntile * 16 + l16;         // B/D: N index for this lane

    v8f c = {};
#pragma unroll
    for (int k0 = 0; k0 < 64; k0 += 4) {
        const int ka = k0 + 2 * hi;
        v2f a, b;
        a.x = X[arow * 64 + ka + 0];
        a.y = X[arow * 64 + ka + 1];
        b.x = W[(ka + 0) * FOUT + ncol];
        b.y = W[(ka + 1) * FOUT + ncol];
        c = __builtin_amdgcn_wmma_f32_16x16x4_f32(
                /*neg_a=*/false, a, /*neg_b=*/false, b,
                /*c_mod=*/(short)0, c, /*reuse_a=*/false, /*reuse_b=*/false);
    }
#pragma unroll
    for (int j = 0; j < 8; ++j) {
        Y[(mtile * 16 + j + hi * 8) * FOUT + ncol] = c[j];
    }
}

// ---------------------------------------------------------------------------
// agg[n,:] = dinv[n]^2 * Y[n,:]         (self-loop contribution, norm = dinv^2)
// ---------------------------------------------------------------------------
template <int F>
__global__ void init_agg_kernel(const float* __restrict__ dinv,
                                const float* __restrict__ Y,
                                float* __restrict__ agg) {
    int i = blockIdx.x * blockDim.x + threadIdx.x;
    if (i < N_NODES * F) {
        float di = dinv[i / F];
        agg[i] = di * di * Y[i];
    }
}

// ---------------------------------------------------------------------------
// Edge scatter: agg[col,:] += dinv[row]*ew*dinv[col] * H[row,:]
// F/4 threads per edge, each handling a float4 feature chunk (coalesced loads,
// hardware f32 global atomics for the adds).
// ---------------------------------------------------------------------------
template <int F>
__global__ void edge_scatter_kernel(const long long* __restrict__ ei,
                                    const float* __restrict__ ew,
                                    const float* __restrict__ dinv,
                                    const float* __restrict__ H,
                                    float* __restrict__ agg) {
    constexpr int G = F / 4;
    int t = blockIdx.x * blockDim.x + threadIdx.x;
    int e = t / G;
    int g = t % G;
    if (e >= N_EDGES) return;

    int r = (int)ei[e];
    int c = (int)ei[N_EDGES + e];
    float nrm = dinv[r] * ew[e] * dinv[c];

    const float4 h = *(const float4*)(H + r * F + 4 * g);
    float* dst = agg + c * F + 4 * g;
    unsafeAtomicAdd(dst + 0, nrm * h.x);
    unsafeAtomicAdd(dst + 1, nrm * h.y);
    unsafeAtomicAdd(dst + 2, nrm * h.z);
    unsafeAtomicAdd(dst + 3, nrm * h.w);
}

// ---------------------------------------------------------------------------
// out[n,f] = act(agg[n,f] + b[f]);  RELU selects the activation.
// ---------------------------------------------------------------------------
template <int F, bool RELU>
__global__ void bias_act_kernel(const float* __restrict__ agg,
                                const float* __restrict__ b,
                                float* __restrict__ out) {
    int i = blockIdx.x * blockDim.x + threadIdx.x;
    if (i < N_NODES * F) {
        float v = agg[i] + b[i % F];
        out[i] = RELU ? fmaxf(v, 0.0f) : v;
    }
}

// ---------------------------------------------------------------------------
extern "C" void kernel_launch(void* const* d_in, const int* in_sizes, int n_in,
                              void* d_out, int out_size, void* d_ws, size_t ws_size,
                              hipStream_t stream) {
    const float*     in_feat = (const float*)d_in[0];
    const long long* ei      = (const long long*)d_in[1];   // int64 edge_index [2, E]
    const float*     ew      = (const float*)d_in[2];
    const float*     W1      = (const float*)d_in[3];
    const float*     b1      = (const float*)d_in[4];
    const float*     W2      = (const float*)d_in[5];
    const float*     b2      = (const float*)d_in[6];
    float*           out     = (float*)d_out;               // [100000, 16]

    // Workspace carve-up (floats): dinv | h (N x 64) | agg (N x 64, reused for z)
    float* ws   = (float*)d_ws;
    float* dinv = ws;                                // 100000
    float* h    = ws + N_NODES;                      // 6,400,000
    float* agg  = ws + N_NODES + N_NODES * 64;       // 6,400,000  (~51.6 MB total)

    const int TB = 256;

    // --- normalization coefficients -------------------------------------
    deg_init_kernel<<<(N_NODES + TB - 1) / TB, TB, 0, stream>>>(dinv);
    deg_scatter_kernel<<<(N_EDGES + TB - 1) / TB, TB, 0, stream>>>(ei, ew, dinv);
    dinv_kernel<<<(N_NODES + TB - 1) / TB, TB, 0, stream>>>(dinv);

    // --- layer 1: h = relu(Anorm @ (x @ W1) + b1) -----------------------
    {
        int waves  = MTILES * 4;                     // 4 N-tiles of 16
        int blocks = waves / 4;                      // 128 threads = 4 waves/block
        gemm_wmma_f32_kernel<64><<<blocks, 128, 0, stream>>>(in_feat, W1, h);
    }
    init_agg_kernel<64><<<(N_NODES * 64 + TB - 1) / TB, TB, 0, stream>>>(dinv, h, agg);
    edge_scatter_kernel<64><<<(N_EDGES * 16 + TB - 1) / TB, TB, 0, stream>>>(ei, ew, dinv, h, agg);
    bias_act_kernel<64, true><<<(N_NODES * 64 + TB - 1) / TB, TB, 0, stream>>>(agg, b1, h);

    // --- layer 2: out = Anorm @ (h @ W2) + b2 ---------------------------
    {
        int waves  = MTILES;                         // 1 N-tile of 16
        int blocks = (waves + 3) / 4;
        gemm_wmma_f32_kernel<16><<<blocks, 128, 0, stream>>>(h, W2, agg); // z -> agg
    }
    init_agg_kernel<16><<<(N_NODES * 16 + TB - 1) / TB, TB, 0, stream>>>(dinv, agg, out);
    edge_scatter_kernel<16><<<(N_EDGES * 4 + TB - 1) / TB, TB, 0, stream>>>(ei, ew, dinv, agg, out);
    bias_act_kernel<16, false><<<(N_NODES * 16 + TB - 1) / TB, TB, 0, stream>>>(out, b2, out);
}